// Layer_23493471109148
// MI455X (gfx1250) — compile-verified
//
#include <hip/hip_runtime.h>
#include <hip/hip_bf16.h>

// SPMM scatter-add: out[dst[e], :] += vals[e] * emb[src[e], :], DIM = 64.
//
// wave32 mapping: one wave per edge; lane handles dims [2*lane, 2*lane+1] as
// a float2 -> one global_load_b64 gather + two non-returning
// global_atomic_add_f32 (relaxed, agent scope -> RMW resolves in L2 atomic
// units; entire 65 MB working set is resident in MI455X's 192 MB L2).
// Edge metadata (src, dst, val) is wave-uniform -> forced into SGPRs via
// readfirstlane so it flows through the scalar memory path (s_load/KMcnt)
// and leaves all VMEM issue slots for the gather + atomics.
// All indexing is int32: max flat offset is 6.4M elements, so 32-bit math
// avoids 64-bit VALU/SALU bloat and enables saddr+voffset addressing.

#define DIM 64

__global__ __launch_bounds__(256) void spmm_scatter_kernel(
    const float* __restrict__ emb,
    const int*   __restrict__ src,
    const int*   __restrict__ dst,
    const float* __restrict__ vals,
    float*       __restrict__ out,
    int n_edges) {
    const int lane = threadIdx.x & 31;
    int edge = (int)((blockIdx.x * blockDim.x + threadIdx.x) >> 5);
    if (edge >= n_edges) return;            // wave-uniform branch

    // Wave-uniform scalarization (wave32): edge metadata lives in SGPRs.
    edge = __builtin_amdgcn_readfirstlane(edge);
    const int s = __builtin_amdgcn_readfirstlane(src[edge]);
    const int d = __builtin_amdgcn_readfirstlane(dst[edge]);
    const float w = __uint_as_float(
        __builtin_amdgcn_readfirstlane(__float_as_uint(vals[edge])));

    // Gather: 32 lanes x float2 = 256B row -> global_load_b64 per lane.
    const float2 v = *(const float2*)(emb + s * DIM + 2 * lane);

    // Scatter-add: non-returning relaxed agent-scope f32 atomics ->
    // global_atomic_add_f32 executed at L2, no pre-op value returned.
    float* o = out + d * DIM + 2 * lane;
    __hip_atomic_fetch_add(o,     v.x * w, __ATOMIC_RELAXED, __HIP_MEMORY_SCOPE_AGENT);
    __hip_atomic_fetch_add(o + 1, v.y * w, __ATOMIC_RELAXED, __HIP_MEMORY_SCOPE_AGENT);
}

__global__ __launch_bounds__(256) void zero4_kernel(float4* __restrict__ out4,
                                                    int n4) {
    int i = blockIdx.x * blockDim.x + threadIdx.x;
    if (i < n4) out4[i] = make_float4(0.f, 0.f, 0.f, 0.f);
}

__global__ __launch_bounds__(64) void zero_tail_kernel(float* __restrict__ out,
                                                       int n) {
    int i = blockIdx.x * blockDim.x + threadIdx.x;
    if (i < n) out[i] = 0.f;
}

extern "C" void kernel_launch(void* const* d_in, const int* in_sizes, int n_in,
                              void* d_out, int out_size, void* d_ws, size_t ws_size,
                              hipStream_t stream) {
    (void)n_in; (void)d_ws; (void)ws_size;

    const float* users_emb = (const float*)d_in[0];
    const float* items_emb = (const float*)d_in[1];
    const int*   user_src  = (const int*)  d_in[2];
    const int*   user_dst  = (const int*)  d_in[3];
    const float* user_vals = (const float*)d_in[4];
    const int*   item_src  = (const int*)  d_in[5];
    const int*   item_dst  = (const int*)  d_in[6];
    const float* item_vals = (const float*)d_in[7];
    // d_in[8..10] (graph_*) are dead in the reference: outputs are (h_u1, h_i1) only.

    float* out = (float*)d_out;
    const int num_user = in_sizes[0] / DIM;   // 60000
    const int e_u = in_sizes[2];              // 720000
    const int e_i = in_sizes[5];              // 480000

    // 1) Zero-init d_out (harness poisons it). b128 stores, scalar tail only
    //    if out_size % 4 != 0 (it is 0 here -> no tail launch).
    {
        int n4 = out_size >> 2;
        if (n4 > 0)
            zero4_kernel<<<(n4 + 255) / 256, 256, 0, stream>>>((float4*)out, n4);
        int tail = out_size & 3;
        if (tail > 0)
            zero_tail_kernel<<<1, 64, 0, stream>>>(out + 4 * n4, tail);
    }

    // 2) User SPMM: one wave (32 lanes) per edge.
    {
        long long threads = (long long)e_u * 32;
        int blocks = (int)((threads + 255) / 256);
        spmm_scatter_kernel<<<blocks, 256, 0, stream>>>(
            users_emb, user_src, user_dst, user_vals, out, e_u);
    }

    // 3) Item SPMM into the h_i1 region of d_out.
    {
        long long threads = (long long)e_i * 32;
        int blocks = (int)((threads + 255) / 256);
        spmm_scatter_kernel<<<blocks, 256, 0, stream>>>(
            items_emb, item_src, item_dst, item_vals,
            out + num_user * DIM, e_i);
    }
}